// GAE_7370163880377
// MI455X (gfx1250) — compile-verified
//
#include <hip/hip_runtime.h>

// GAE backward scan, MI455X (gfx1250, wave32).
// Layout: (B=8192, T=256, A=4, 1) fp32. One wave per batch row; lane l owns
// timesteps [8l, 8l+8); agents ride in a float4 lane. Affine-map suffix scan
// across lanes via __shfl_down (5 steps), then local replay. Single pass over
// memory: 201 MB total -> ~8.6 us floor at 23.3 TB/s.

typedef float v4f __attribute__((ext_vector_type(4)));

#define GAMMA 0.99f
#define LMBDA 0.95f

constexpr int T = 256;     // timesteps
constexpr int A = 4;       // agents (float4 lane)
constexpr int CHUNK = 8;   // T / 32 lanes

__device__ __forceinline__ v4f v4_splat(float s) {
    v4f r; r.x = s; r.y = s; r.z = s; r.w = s; return r;
}

__device__ __forceinline__ v4f shfl_down_v4(v4f x, int delta) {
    v4f r;
    r.x = __shfl_down(x.x, delta, 32);
    r.y = __shfl_down(x.y, delta, 32);
    r.z = __shfl_down(x.z, delta, 32);
    r.w = __shfl_down(x.w, delta, 32);
    return r;
}

__global__ void __launch_bounds__(256)
gae_wave_scan(const float* __restrict__ reward,
              const float* __restrict__ terminated,
              const float* __restrict__ value,
              const float* __restrict__ next_value,
              float* __restrict__ out,   // [adv (B*T*A) | ret (B*T*A)]
              int B) {
    const int gtid = blockIdx.x * blockDim.x + threadIdx.x;
    const int b    = gtid >> 5;          // one wave32 per batch row
    const int lane = threadIdx.x & 31;
    if (b >= B) return;

    const long base = (long)b * T + (long)lane * CHUNK;   // in float4 units
    const v4f* R  = (const v4f*)reward     + base;
    const v4f* Td = (const v4f*)terminated + base;
    const v4f* V  = (const v4f*)value      + base;
    const v4f* NV = (const v4f*)next_value + base;

    // gfx1250 global_prefetch_b8 for this lane's tile
    __builtin_prefetch((const void*)R,  0, 0);
    __builtin_prefetch((const void*)Td, 0, 0);
    __builtin_prefetch((const void*)V,  0, 0);
    __builtin_prefetch((const void*)NV, 0, 0);

    v4f dlt[CHUNK], cc[CHUNK], vv[CHUNK];
#pragma unroll
    for (int j = 0; j < CHUNK; ++j) {
        v4f r  = __builtin_nontemporal_load(R  + j);
        v4f td = __builtin_nontemporal_load(Td + j);
        v4f v  = __builtin_nontemporal_load(V  + j);
        v4f nv = __builtin_nontemporal_load(NV + j);
        v4f nd = 1.0f - td;
        dlt[j] = r + GAMMA * nv * nd - v;   // delta_t
        cc[j]  = (GAMMA * LMBDA) * nd;      // recurrence coefficient
        vv[j]  = v;                         // kept for returns = adv + value
    }

    // Compose this lane's 8 steps (backward in time) into one affine map:
    //   g_out = Dacc + Cacc * g_in   (g_in = gae entering from later time)
    v4f Cacc = v4_splat(1.0f);
    v4f Dacc = v4_splat(0.0f);
#pragma unroll
    for (int j = CHUNK - 1; j >= 0; --j) {
        Dacc = dlt[j] + cc[j] * Dacc;
        Cacc = Cacc * cc[j];
    }

    // Inclusive suffix scan of affine maps across lanes (Hillis-Steele).
    // Lane l ends holding M_l o M_{l+1} o ... o M_31.
#pragma unroll
    for (int s = 1; s < 32; s <<= 1) {
        v4f oC = shfl_down_v4(Cacc, s);
        v4f oD = shfl_down_v4(Dacc, s);
        if (lane + s < 32) {
            Dacc = Dacc + Cacc * oD;   // own o other  (other = later time)
            Cacc = Cacc * oC;
        }
    }

    // Incoming gae state for this lane = suffix of lane+1 applied to 0.
    v4f gin = shfl_down_v4(Dacc, 1);
    if (lane == 31) gin = v4_splat(0.0f);

    // Replay local chunk with the true incoming state.
    v4f adv[CHUNK];
    v4f g = gin;
#pragma unroll
    for (int j = CHUNK - 1; j >= 0; --j) {
        g = dlt[j] + cc[j] * g;
        adv[j] = g;
    }

    v4f* Oa = (v4f*)out + base;                      // advantages
    v4f* Or = (v4f*)out + (long)B * T + base;        // returns
#pragma unroll
    for (int j = 0; j < CHUNK; ++j) {
        __builtin_nontemporal_store(adv[j],         Oa + j);
        __builtin_nontemporal_store(adv[j] + vv[j], Or + j);
    }
}

extern "C" void kernel_launch(void* const* d_in, const int* in_sizes, int n_in,
                              void* d_out, int out_size, void* d_ws, size_t ws_size,
                              hipStream_t stream) {
    const float* reward     = (const float*)d_in[0];
    const float* terminated = (const float*)d_in[1];
    const float* value      = (const float*)d_in[2];
    const float* next_value = (const float*)d_in[3];
    float* out = (float*)d_out;

    const int B = in_sizes[0] / (T * A);             // 8192
    const int threads = 256;                         // 8 waves per block
    const int blocks  = (B * 32 + threads - 1) / threads;

    gae_wave_scan<<<blocks, threads, 0, stream>>>(reward, terminated, value,
                                                  next_value, out, B);
}